// TransitionDown_34926674051782
// MI455X (gfx1250) — compile-verified
//
#include <hip/hip_runtime.h>
#include <float.h>

// Problem constants (B derived at launch from in_sizes).
#define NPTS 4096   // input points per batch
#define MPTS 1024   // FPS-sampled points
#define KNN  16     // neighbors
#define CIN  256    // feature dim
#define COUT 512    // output dim (2C)
#define A_STRIDE 260 // 256 + 4 pad -> conflict-free LDS column reads (64 banks)
#define ROWS_PER_BLOCK 32
#define GEMM_BM 128  // GEMM rows per block (2 x 16-row tiles per wave, 4 wave rows)
#define GEMM_BN 128  // GEMM cols per block (4 x 16-col tiles per wave, 2 wave cols)

typedef __attribute__((ext_vector_type(2))) float v2f;
typedef __attribute__((ext_vector_type(8))) float v8f;

__device__ __forceinline__ void argmax_combine(float& v, int& i, float ov, int oi) {
  if (ov > v || (ov == v && oi < i)) { v = ov; i = oi; }
}
__device__ __forceinline__ void argmin_combine(float& v, int& i, float ov, int oi) {
  if (ov < v || (ov == v && oi < i)) { v = ov; i = oi; }
}

// ---------------------------------------------------------------------------
// Kernel 1: farthest point sampling. One block per batch, 1024 threads
// (32 wave32 waves). Coords staged in LDS; per-thread running min-dists in
// registers; block argmax via wave shuffles + LDS combine.
// ---------------------------------------------------------------------------
__global__ __launch_bounds__(1024) void fps_kernel(const float* __restrict__ coords,
                                                   float* __restrict__ new_coords) {
  __shared__ float cx[NPTS], cy[NPTS], cz[NPTS];
  __shared__ float rv[32];
  __shared__ int   ri[32];
  __shared__ int   s_far;

  const int b    = blockIdx.x;
  const int tid  = threadIdx.x;
  const int lane = tid & 31;
  const int wid  = tid >> 5;

  const float* cb = coords + (size_t)b * NPTS * 3;
  for (int i = tid; i < NPTS; i += 1024) {
    cx[i] = cb[i * 3 + 0];
    cy[i] = cb[i * 3 + 1];
    cz[i] = cb[i * 3 + 2];
  }
  if (tid == 0) s_far = 0;
  __syncthreads();

  float dists[4] = {FLT_MAX, FLT_MAX, FLT_MAX, FLT_MAX};
  int far = 0;

  for (int it = 0; it < MPTS; ++it) {
    const float px = cx[far], py = cy[far], pz = cz[far];
    if (tid == 0) {
      float* o = new_coords + ((size_t)b * MPTS + it) * 3;
      o[0] = px; o[1] = py; o[2] = pz;
    }

    float best = -FLT_MAX;
    int   bidx = 0;
#pragma unroll
    for (int j = 0; j < 4; ++j) {
      const int p = tid + j * 1024;
      const float dx = cx[p] - px, dy = cy[p] - py, dz = cz[p] - pz;
      const float d  = dx * dx + dy * dy + dz * dz;
      const float nd = fminf(dists[j], d);
      dists[j] = nd;
      if (nd > best || (nd == best && p < bidx)) { best = nd; bidx = p; }
    }

#pragma unroll
    for (int off = 16; off >= 1; off >>= 1) {
      const float ov = __shfl_xor(best, off, 32);
      const int   oi = __shfl_xor(bidx, off, 32);
      argmax_combine(best, bidx, ov, oi);
    }
    if (lane == 0) { rv[wid] = best; ri[wid] = bidx; }
    __syncthreads();

    if (wid == 0) {
      best = rv[lane];
      bidx = ri[lane];
#pragma unroll
      for (int off = 16; off >= 1; off >>= 1) {
        const float ov = __shfl_xor(best, off, 32);
        const int   oi = __shfl_xor(bidx, off, 32);
        argmax_combine(best, bidx, ov, oi);
      }
      if (lane == 0) s_far = bidx;
    }
    __syncthreads();
    far = s_far;
  }
}

// ---------------------------------------------------------------------------
// Kernel 2: KNN among the sampled points. Each block handles ROWS_PER_BLOCK
// query rows of one batch; sampled points + d2 row live in LDS; 16 iterative
// block-argmins with lowest-index tie-break (matches top_k semantics).
// ---------------------------------------------------------------------------
__global__ __launch_bounds__(256) void knn_kernel(const float* __restrict__ spts,
                                                  int* __restrict__ knn_idx) {
  __shared__ float sx[MPTS], sy[MPTS], sz[MPTS];
  __shared__ float d2[MPTS];
  __shared__ float rv[8];
  __shared__ int   ri[8];

  const int b    = blockIdx.y;
  const int r0   = blockIdx.x * ROWS_PER_BLOCK;
  const int tid  = threadIdx.x;
  const int lane = tid & 31;
  const int wid  = tid >> 5;

  const float* pb = spts + (size_t)b * MPTS * 3;
  for (int i = tid; i < MPTS; i += 256) {
    sx[i] = pb[i * 3 + 0];
    sy[i] = pb[i * 3 + 1];
    sz[i] = pb[i * 3 + 2];
  }
  __syncthreads();

  for (int rr = 0; rr < ROWS_PER_BLOCK; ++rr) {
    const int row = r0 + rr;
    const float qx = sx[row], qy = sy[row], qz = sz[row];
    for (int i = tid; i < MPTS; i += 256) {
      const float dx = sx[i] - qx, dy = sy[i] - qy, dz = sz[i] - qz;
      d2[i] = dx * dx + dy * dy + dz * dz;
    }
    __syncthreads();

    for (int t = 0; t < KNN; ++t) {
      float best = FLT_MAX;
      int   bidx = MPTS;
      for (int i = tid; i < MPTS; i += 256) {
        const float v = d2[i];
        if (v < best || (v == best && i < bidx)) { best = v; bidx = i; }
      }
#pragma unroll
      for (int off = 16; off >= 1; off >>= 1) {
        const float ov = __shfl_xor(best, off, 32);
        const int   oi = __shfl_xor(bidx, off, 32);
        argmin_combine(best, bidx, ov, oi);
      }
      if (lane == 0) { rv[wid] = best; ri[wid] = bidx; }
      __syncthreads();

      if (wid == 0) {
        float v  = (lane < 8) ? rv[lane] : FLT_MAX;
        int   ix = (lane < 8) ? ri[lane] : MPTS;
#pragma unroll
        for (int off = 4; off >= 1; off >>= 1) {
          const float ov = __shfl_xor(v, off, 32);
          const int   oi = __shfl_xor(ix, off, 32);
          argmin_combine(v, ix, ov, oi);
        }
        if (lane == 0) {
          knn_idx[((size_t)b * MPTS + row) * KNN + t] = ix;
          d2[ix] = FLT_MAX;  // remove from candidate set
        }
      }
      __syncthreads();
    }
  }
}

// ---------------------------------------------------------------------------
// Kernel 3a: H[b,j,:] = relu(features[b,j,:] @ W + bias), j in [0, MPTS).
// Only rows < MPTS are ever gathered by the reference, so the linear layer is
// computed ONCE per unique row (16x fewer FLOPs than per-neighbor).
// Dense GEMM: (B*MPTS x CIN) @ (CIN x COUT) with v_wmma_f32_16x16x4_f32.
// Block = 128 rows x 128 cols; 8 waves in 4(row) x 2(col) arrangement; each
// wave owns a 32x64 strip = 2 row-tiles x 4 col-tiles (8 accumulators).
// Each B fragment feeds TWO WMMAs (shared across the wave's row tiles),
// halving the global-load : WMMA ratio vs a 16-row wave tile.
// ---------------------------------------------------------------------------
__global__ __launch_bounds__(256) void linear_relu_kernel(
    const float* __restrict__ features, const float* __restrict__ W,
    const float* __restrict__ bias, float* __restrict__ H) {
  __shared__ __align__(16) float As[GEMM_BM * A_STRIDE];  // 128x260 = 133 KB

  const int tid   = threadIdx.x;
  const int lane  = tid & 31;
  const int wid   = tid >> 5;
  const int row0  = blockIdx.y * GEMM_BM;   // global row base in [0, B*MPTS)
  const int ncol0 = blockIdx.x * GEMM_BN;   // column base

  // Stage A tile (128 rows x 256 K) via float4; rows map to features[b, j<MPTS].
  for (int e = tid; e < GEMM_BM * (CIN / 4); e += 256) {
    const int r    = e >> 6;                 // CIN/4 == 64
    const int c4   = e & 63;
    const int grow = row0 + r;
    const int b    = grow >> 10;             // / MPTS
    const int j    = grow & (MPTS - 1);
    const float4 v =
        *(const float4*)(features + ((size_t)b * NPTS + j) * CIN + c4 * 4);
    *(float4*)(As + r * A_STRIDE + c4 * 4) = v;
  }
  __syncthreads();

  // WMMA f32 16x16x4 operand mapping (wave32):
  //   A frag: lane l holds A[l%16][kk + 2*(l/16) + {0,1}]
  //   B frag: lane l holds W[kk + 2*(l/16) + {0,1}][n0 + l%16]
  const int wr    = wid & 3;                 // wave row group: rows 32*wr
  const int wc    = wid >> 2;                // wave col group: cols 64*wc
  const int col   = lane & 15;
  const int khalf = (lane >> 4) * 2;         // 0 or 2
  const float* arow0 = As + (wr * 32 + col) * A_STRIDE + khalf;
  const float* arow1 = arow0 + 16 * A_STRIDE;
  const int    nbase = ncol0 + wc * 64;

  v8f acc[8];
#pragma unroll
  for (int t = 0; t < 8; ++t) acc[t] = (v8f){0.f,0.f,0.f,0.f,0.f,0.f,0.f,0.f};

  const float* wp = W + (size_t)khalf * COUT + nbase + col;
#pragma unroll 4
  for (int kk = 0; kk < CIN; kk += 4) {
    v2f a0, a1;
    a0.x = arow0[kk + 0];
    a0.y = arow0[kk + 1];
    a1.x = arow1[kk + 0];
    a1.y = arow1[kk + 1];
#pragma unroll
    for (int t = 0; t < 4; ++t) {
      v2f bf;
      bf.x = wp[(size_t)(kk + 0) * COUT + t * 16];
      bf.y = wp[(size_t)(kk + 1) * COUT + t * 16];
      acc[t] = __builtin_amdgcn_wmma_f32_16x16x4_f32(
          false, a0, false, bf, (short)0, acc[t], false, false);
      acc[4 + t] = __builtin_amdgcn_wmma_f32_16x16x4_f32(
          false, a1, false, bf, (short)0, acc[4 + t], false, false);
    }
  }

  // Epilogue: bias + relu; D layout VGPR j -> rows M=j (lanes 0-15) / M=j+8.
  const int rbase0 = row0 + wr * 32 + (lane >> 4) * 8;
#pragma unroll
  for (int t = 0; t < 4; ++t) {
    const int n  = nbase + t * 16 + col;
    const float bv = bias[n];
#pragma unroll
    for (int j = 0; j < 8; ++j) {
      H[(size_t)(rbase0 + j) * COUT + n]      = fmaxf(acc[t][j] + bv, 0.f);
      H[(size_t)(rbase0 + 16 + j) * COUT + n] = fmaxf(acc[4 + t][j] + bv, 0.f);
    }
  }
}

// ---------------------------------------------------------------------------
// Kernel 3b: agg[b,m,:] = max over 16 neighbors of H rows (pure gather-max;
// H is 33.5 MB -> fully L2-resident on the 192 MB L2).
// ---------------------------------------------------------------------------
__global__ __launch_bounds__(256) void gather_max_kernel(
    const float* __restrict__ H, const int* __restrict__ knn_idx,
    float* __restrict__ agg) {
  __shared__ int nidx[KNN];
  const int b   = blockIdx.y;
  const int m   = blockIdx.x;
  const int tid = threadIdx.x;
  const size_t bm = (size_t)b * MPTS + m;
  if (tid < KNN) nidx[tid] = knn_idx[bm * KNN + tid];
  __syncthreads();

  const int n = tid * 2;  // 256 threads x float2 = 512 cols
  float2 mx = make_float2(-FLT_MAX, -FLT_MAX);
#pragma unroll
  for (int t = 0; t < KNN; ++t) {
    const float2 v =
        *(const float2*)(H + ((size_t)b * MPTS + nidx[t]) * COUT + n);
    mx.x = fmaxf(mx.x, v.x);
    mx.y = fmaxf(mx.y, v.y);
  }
  *(float2*)(agg + bm * COUT + n) = mx;
}

// ---------------------------------------------------------------------------
// Fallback (used only if d_ws is too small for H): fused per-(b,m) gather +
// GEMM + relu + max. Also WMMA-based.
// ---------------------------------------------------------------------------
__global__ __launch_bounds__(256) void gather_gemm_max_kernel(
    const float* __restrict__ features, const float* __restrict__ W,
    const float* __restrict__ bias, const int* __restrict__ knn_idx,
    float* __restrict__ agg) {
  __shared__ float As[KNN * A_STRIDE];
  __shared__ int   nidx[KNN];

  const int b    = blockIdx.y;
  const int m    = blockIdx.x;
  const int tid  = threadIdx.x;
  const int lane = tid & 31;
  const int wid  = tid >> 5;
  const size_t bm = (size_t)b * MPTS + m;

  if (tid < KNN) nidx[tid] = knn_idx[bm * KNN + tid];
  __syncthreads();

  for (int e = tid; e < KNN * CIN; e += 256) {
    const int r = e >> 8;
    const int c = e & (CIN - 1);
    As[r * A_STRIDE + c] = features[((size_t)b * NPTS + nidx[r]) * CIN + c];
  }
  __syncthreads();

  const int col   = lane & 15;
  const int khalf = (lane >> 4) * 2;
  const float* arow = As + col * A_STRIDE + khalf;

#pragma unroll
  for (int t = 0; t < 4; ++t) {
    const int n0 = (wid * 4 + t) * 16;
    v8f acc = {0.f, 0.f, 0.f, 0.f, 0.f, 0.f, 0.f, 0.f};
    const float* wp = W + (size_t)khalf * COUT + n0 + col;

#pragma unroll 4
    for (int kk = 0; kk < CIN; kk += 4) {
      v2f a;
      a.x = arow[kk + 0];
      a.y = arow[kk + 1];
      v2f bf;
      bf.x = wp[(size_t)(kk + 0) * COUT];
      bf.y = wp[(size_t)(kk + 1) * COUT];
      acc = __builtin_amdgcn_wmma_f32_16x16x4_f32(
          false, a, false, bf, (short)0, acc, false, false);
    }

    float mx = acc[0];
#pragma unroll
    for (int j = 1; j < 8; ++j) mx = fmaxf(mx, acc[j]);
    mx = fmaxf(mx, __shfl_xor(mx, 16, 32));
    const float val = fmaxf(mx + bias[n0 + col], 0.f);
    if (lane < 16) agg[bm * COUT + n0 + col] = val;
  }
}

// ---------------------------------------------------------------------------
extern "C" void kernel_launch(void* const* d_in, const int* in_sizes, int n_in,
                              void* d_out, int out_size, void* d_ws, size_t ws_size,
                              hipStream_t stream) {
  const float* coords   = (const float*)d_in[0];  // (B, 4096, 3)
  const float* features = (const float*)d_in[1];  // (B, 4096, 256)
  const float* W        = (const float*)d_in[2];  // (256, 512)
  const float* bias     = (const float*)d_in[3];  // (512,)
  float* out = (float*)d_out;

  const int B = in_sizes[0] / (NPTS * 3);

  float* new_coords = out;                          // B*M*3 floats
  float* agg        = out + (size_t)B * MPTS * 3;   // B*M*COUT floats

  // Workspace layout: knn indices, then H (precomputed linear+relu rows).
  int*   knn_idx   = (int*)d_ws;
  const size_t knn_bytes = (size_t)B * MPTS * KNN * sizeof(int);
  const size_t h_off     = (knn_bytes + 255) & ~(size_t)255;
  const size_t need      = h_off + (size_t)B * MPTS * COUT * sizeof(float);

  fps_kernel<<<B, 1024, 0, stream>>>(coords, new_coords);
  knn_kernel<<<dim3(MPTS / ROWS_PER_BLOCK, B), 256, 0, stream>>>(new_coords,
                                                                 knn_idx);

  if (ws_size >= need) {
    float* H = (float*)((char*)d_ws + h_off);
    linear_relu_kernel<<<dim3(COUT / GEMM_BN, (B * MPTS) / GEMM_BM), 256, 0,
                         stream>>>(features, W, bias, H);
    gather_max_kernel<<<dim3(MPTS, B), 256, 0, stream>>>(H, knn_idx, agg);
  } else {
    gather_gemm_max_kernel<<<dim3(MPTS, B), 256, 0, stream>>>(features, W, bias,
                                                              knn_idx, agg);
  }
}